// IGNet_34720515621701
// MI455X (gfx1250) — compile-verified
//
#include <hip/hip_runtime.h>
#include <hip/hip_bf16.h>

#define NV    300
#define NA    12
#define ND    4
#define NM    3600      // NV*NA
#define NSEED 1024
#define NGP   1024
#define NBATCH 2
#define PI_F  3.14159265358979323846

typedef __attribute__((ext_vector_type(2))) float v2f;
typedef __attribute__((ext_vector_type(8))) float v8f;

// ---------------- workspace layout (float units) ----------------
#define OF_VIEWS   0
#define OF_GROT    (OF_VIEWS + NV*3)        // 3600 x 9
#define OF_PKP     (OF_GROT + NM*9)         // 3600 x 12
#define OF_BATCH   (OF_PKP + NM*12)
// per-batch block
#define PB_VIND    0                        // int[300]
#define PB_AIND    (PB_VIND + NV)           // int[3600]
#define PB_TR      (PB_AIND + NM)           // 3600 x 9
#define PB_TB      (PB_TR + NM*9)           // 3600 x 12
#define PB_TBS     (PB_TB + NM*12)          // 3600 x 12
#define PB_TN1     (PB_TBS + NM*12)         // 3600
#define PB_TN2     (PB_TN1 + NM)            // 3600
#define PB_GPT     (PB_TN2 + NM)            // 1024 x 3
#define PB_NN      (PB_GPT + NGP*3)         // int[1024]
#define PB_PBASE   (PB_NN + NSEED)          // 1024 x 12
#define PB_PNORM   (PB_PBASE + NSEED*12)    // 1024
#define PB_RIND    (PB_PNORM + NSEED)       // int[1024]
#define PB_SIZE    (PB_RIND + NSEED)

__device__ const float c_local[4][3] = {
    {0.02f, 0.0f, 0.0f}, {0.02f, 0.01f, 0.0f}, {0.02f, -0.01f, 0.0f}, {0.0f, 0.0f, 0.0f}};

__device__ inline void load_R(const float* pose, int b, float* R) {
    const float* P = pose + b * 12;
    for (int i = 0; i < 3; i++)
        for (int j = 0; j < 3; j++) R[i * 3 + j] = P[i * 4 + j];
}

// ---------------- K0: constant tables (VIEWS, GRASP_ROT, PKP) ----------------
__global__ void k0_const(float* ws) {
    int m = blockIdx.x * blockDim.x + threadIdx.x;
    if (m >= NM) return;
    int v = m / NA, a = m % NA;
    double phi = (sqrt(5.0) - 1.0) / 2.0;
    double z = (2.0 * v + 1.0) / (double)NV - 1.0;
    double r = sqrt(fmax(1.0 - z * z, 0.0));
    double th = 2.0 * PI_F * (double)v * phi;
    float vx = (float)(r * cos(th)), vy = (float)(r * sin(th)), vz = (float)z;
    if (a == 0) {
        float* V = ws + OF_VIEWS + v * 3;
        V[0] = vx; V[1] = vy; V[2] = vz;
    }
    // towards = -view (already unit length)
    float ax[3] = {-vx, -vy, -vz};
    float ay[3] = {-ax[1], ax[0], 0.0f};
    float n = sqrtf(ay[0] * ay[0] + ay[1] * ay[1]);
    if (n > 1e-8f) {
        float inv = 1.0f / fmaxf(n, 1e-12f);
        ay[0] *= inv; ay[1] *= inv; ay[2] *= inv;
    } else { ay[0] = 0.0f; ay[1] = 1.0f; ay[2] = 0.0f; }
    float az[3] = {ax[1] * ay[2] - ax[2] * ay[1],
                   ax[2] * ay[0] - ax[0] * ay[2],
                   ax[0] * ay[1] - ax[1] * ay[0]};
    float ang = (float)(PI_F / NA) * (float)a;
    float c = cosf(ang), s = sinf(ang);
    float G[9];
    for (int i = 0; i < 3; i++) {      // R2 = [ax ay az] cols, then @ rotX(ang)
        G[i * 3 + 0] = ax[i];
        G[i * 3 + 1] = ay[i] * c + az[i] * s;
        G[i * 3 + 2] = -ay[i] * s + az[i] * c;
    }
    float* Gp = ws + OF_GROT + m * 9;
    for (int i = 0; i < 9; i++) Gp[i] = G[i];
    float* P = ws + OF_PKP + m * 12;
    for (int k = 0; k < 4; k++)
        for (int i = 0; i < 3; i++)
            P[k * 3 + i] = G[i * 3] * c_local[k][0] + G[i * 3 + 1] * c_local[k][1] +
                           G[i * 3 + 2] * c_local[k][2];
}

// ---------------- K1: per-batch view_inds ----------------
__global__ void k1_viewinds(const float* pose, float* ws) {
    int b = blockIdx.x;
    int v = threadIdx.x;
    __shared__ float vt[NV][3];
    float R[9]; load_R(pose, b, R);
    const float* V = ws + OF_VIEWS;
    if (v < NV) {
        float x = V[v * 3], y = V[v * 3 + 1], z = V[v * 3 + 2];
        vt[v][0] = R[0] * x + R[1] * y + R[2] * z;
        vt[v][1] = R[3] * x + R[4] * y + R[5] * z;
        vt[v][2] = R[6] * x + R[7] * y + R[8] * z;
    }
    __syncthreads();
    if (v >= NV) return;
    float ax = V[v * 3], ay = V[v * 3 + 1], az = V[v * 3 + 2];
    float an = ax * ax + ay * ay + az * az;
    float best = 3.4e38f; int bi = 0;
    for (int j = 0; j < NV; j++) {
        float bx = vt[j][0], by = vt[j][1], bz = vt[j][2];
        float d = an + bx * bx + by * by + bz * bz - 2.0f * (ax * bx + ay * by + az * bz);
        if (d < best) { best = d; bi = j; }
    }
    ((int*)(ws + OF_BATCH + (size_t)b * PB_SIZE + PB_VIND))[v] = bi;
}

// ---------------- K2: angle alignment -> aind, tR, tbase, tbase_s, norms ----------------
__global__ void k2_align(const float* pose, float* ws) {
    int b = blockIdx.y;
    int m = blockIdx.x * blockDim.x + threadIdx.x;
    if (m >= NM) return;
    int v = m / NA;
    float R[9]; load_R(pose, b, R);
    float* wb = ws + OF_BATCH + (size_t)b * PB_SIZE;
    int vi = ((const int*)(wb + PB_VIND))[v];
    const float* pk = ws + OF_PKP + m * 12;
    float best1 = 3.4e38f, best2 = 3.4e38f;
    int a1 = 0, a2 = 0;
    for (int ap = 0; ap < NA; ap++) {
        const float* G = ws + OF_GROT + (size_t)(vi * NA + ap) * 9;
        float T[9];
        for (int i = 0; i < 3; i++)
            for (int j = 0; j < 3; j++)
                T[i * 3 + j] = R[i * 3] * G[j] + R[i * 3 + 1] * G[3 + j] + R[i * 3 + 2] * G[6 + j];
        float d1 = 0.0f, d2 = 0.0f;
        for (int k = 0; k < 4; k++) {
            float lx = c_local[k][0], ly = c_local[k][1], lz = c_local[k][2];
            for (int i = 0; i < 3; i++) {
                float t = T[i * 3] * lx + T[i * 3 + 1] * ly + T[i * 3 + 2] * lz;
                float ts = T[i * 3] * lx - T[i * 3 + 1] * ly + T[i * 3 + 2] * lz;
                float p = pk[k * 3 + i];
                float e1 = p - t, e2 = p - ts;
                d1 += e1 * e1; d2 += e2 * e2;
            }
        }
        if (d1 < best1) { best1 = d1; a1 = ap; }
        if (d2 < best2) { best2 = d2; a2 = ap; }
    }
    int sel = (best1 < best2) ? a1 : a2;
    ((int*)(wb + PB_AIND))[m] = sel;
    const float* G = ws + OF_GROT + (size_t)(vi * NA + sel) * 9;
    float T[9];
    for (int i = 0; i < 3; i++)
        for (int j = 0; j < 3; j++)
            T[i * 3 + j] = R[i * 3] * G[j] + R[i * 3 + 1] * G[3 + j] + R[i * 3 + 2] * G[6 + j];
    float* tR = wb + PB_TR + (size_t)m * 9;
    for (int i = 0; i < 9; i++) tR[i] = T[i];
    float* tb = wb + PB_TB + (size_t)m * 12;
    float* tbs = wb + PB_TBS + (size_t)m * 12;
    float n1 = 0.0f, n2 = 0.0f;
    for (int k = 0; k < 4; k++) {
        float lx = c_local[k][0], ly = c_local[k][1], lz = c_local[k][2];
        for (int i = 0; i < 3; i++) {
            float t = T[i * 3] * lx + T[i * 3 + 1] * ly + T[i * 3 + 2] * lz;
            float ts = T[i * 3] * lx - T[i * 3 + 1] * ly + T[i * 3 + 2] * lz;
            tb[k * 3 + i] = t; tbs[k * 3 + i] = ts;
            n1 += t * t; n2 += ts * ts;
        }
    }
    wb[PB_TN1 + m] = n1;
    wb[PB_TN2 + m] = n2;
}

// ---------------- K3: gp_trans ----------------
__global__ void k3_gpt(const float* pose, const float* gp, float* ws) {
    int b = blockIdx.y;
    int i = blockIdx.x * blockDim.x + threadIdx.x;
    if (i >= NGP) return;
    const float* P = pose + b * 12;
    const float* g = gp + (size_t)(b * NGP + i) * 3;
    float* o = ws + OF_BATCH + (size_t)b * PB_SIZE + PB_GPT + i * 3;
    for (int r = 0; r < 3; r++)
        o[r] = P[r * 4] * g[0] + P[r * 4 + 1] * g[1] + P[r * 4 + 2] * g[2] + P[r * 4 + 3];
}

// ---------------- K4: seed -> nearest grasp point ----------------
__global__ void k4_nn(const float* pc, float* ws) {
    int b = blockIdx.y;
    int n = blockIdx.x * blockDim.x + threadIdx.x;
    if (n >= NSEED) return;
    const float* s = pc + (size_t)(b * NSEED + n) * 3;
    float sx = s[0], sy = s[1], sz = s[2];
    float sn = sx * sx + sy * sy + sz * sz;
    float* wb = ws + OF_BATCH + (size_t)b * PB_SIZE;
    const float* gpt = wb + PB_GPT;
    float best = 3.4e38f; int bi = 0;
    for (int j = 0; j < NGP; j++) {
        float gx = gpt[j * 3], gy = gpt[j * 3 + 1], gz = gpt[j * 3 + 2];
        float d = sn + gx * gx + gy * gy + gz * gz - 2.0f * (sx * gx + sy * gy + sz * gz);
        if (d < best) { best = d; bi = j; }
    }
    ((int*)(wb + PB_NN))[n] = bi;
}

// ---------------- K5: rot6d -> pbase keypoints + norm ----------------
__global__ void k5_pbase(const float* r6in, float* ws) {
    int b = blockIdx.y;
    int n = blockIdx.x * blockDim.x + threadIdx.x;
    if (n >= NSEED) return;
    const float* r6 = r6in + (size_t)(b * NSEED + n) * 6;
    float a1[3] = {r6[0], r6[1], r6[2]}, a2[3] = {r6[3], r6[4], r6[5]};
    float l1 = sqrtf(a1[0] * a1[0] + a1[1] * a1[1] + a1[2] * a1[2]);
    float b1[3] = {a1[0] / l1, a1[1] / l1, a1[2] / l1};
    float dp = b1[0] * a2[0] + b1[1] * a2[1] + b1[2] * a2[2];
    float a2p[3] = {a2[0] - dp * b1[0], a2[1] - dp * b1[1], a2[2] - dp * b1[2]};
    float l2 = sqrtf(a2p[0] * a2p[0] + a2p[1] * a2p[1] + a2p[2] * a2p[2]);
    float b2[3] = {a2p[0] / l2, a2p[1] / l2, a2p[2] / l2};
    float b3[3] = {b1[1] * b2[2] - b1[2] * b2[1],
                   b1[2] * b2[0] - b1[0] * b2[2],
                   b1[0] * b2[1] - b1[1] * b2[0]};
    float Rp[9] = {b1[0], b1[1], b1[2], b2[0], b2[1], b2[2], b3[0], b3[1], b3[2]};
    float* wb = ws + OF_BATCH + (size_t)b * PB_SIZE;
    float* pb = wb + PB_PBASE + (size_t)n * 12;
    float nrm = 0.0f;
    for (int k = 0; k < 4; k++)
        for (int i = 0; i < 3; i++) {
            float p = Rp[i * 3] * c_local[k][0] + Rp[i * 3 + 1] * c_local[k][1] +
                      Rp[i * 3 + 2] * c_local[k][2];
            pb[k * 3 + i] = p;
            nrm += p * p;
        }
    wb[PB_PNORM + n] = nrm;
}

// ---------------- K6: WMMA distance GEMM (K=12 via 3x 16x16x4 f32) + fused argmin ----------------
__global__ void __launch_bounds__(32) k6_argmin(float* ws) {
    int b = blockIdx.y;
    float* wb = ws + OF_BATCH + (size_t)b * PB_SIZE;
    int lane = threadIdx.x;
    int row = lane & 15, half = lane >> 4;
    int seed0 = blockIdx.x * 16;
    const float* pb = wb + PB_PBASE;
    const float* tb = wb + PB_TB;
    const float* tbs = wb + PB_TBS;
    const float* tn1 = wb + PB_TN1;
    const float* tn2 = wb + PB_TN2;
    const float* pn = wb + PB_PNORM;

    // A tile: 16x12 f32 of pbase. ISA 32-bit A layout: lane=M, K halves split on
    // lane half; VGPR kk holds K = 4*kk + 2*half + {0,1}.
    const float* arow = pb + (size_t)(seed0 + row) * 12 + 2 * half;
    v2f A0 = *(const v2f*)(arow + 0);
    v2f A1 = *(const v2f*)(arow + 4);
    v2f A2 = *(const v2f*)(arow + 8);
    float pnr[8];
    for (int r = 0; r < 8; r++) pnr[r] = pn[seed0 + r + 8 * half];

    float mv1[8], mv2[8]; int mi1[8], mi2[8];
    for (int r = 0; r < 8; r++) { mv1[r] = 3.4e38f; mv2[r] = 3.4e38f; mi1[r] = 0; mi2[r] = 0; }

    for (int c0 = 0; c0 < NM; c0 += 16) {
        int ncol = c0 + row;  // C-layout: this lane's N index (both halves)
        const float* br = tb + (size_t)ncol * 12 + 2 * half;
        const float* bs = tbs + (size_t)ncol * 12 + 2 * half;
        v2f B0 = *(const v2f*)(br + 0);
        v2f B1 = *(const v2f*)(br + 4);
        v2f B2 = *(const v2f*)(br + 8);
        v2f S0 = *(const v2f*)(bs + 0);
        v2f S1 = *(const v2f*)(bs + 4);
        v2f S2 = *(const v2f*)(bs + 8);
        v8f acc1 = {};
        acc1 = __builtin_amdgcn_wmma_f32_16x16x4_f32(false, A0, false, B0, (short)0, acc1, false, false);
        acc1 = __builtin_amdgcn_wmma_f32_16x16x4_f32(false, A1, false, B1, (short)0, acc1, false, false);
        acc1 = __builtin_amdgcn_wmma_f32_16x16x4_f32(false, A2, false, B2, (short)0, acc1, false, false);
        v8f acc2 = {};
        acc2 = __builtin_amdgcn_wmma_f32_16x16x4_f32(false, A0, false, S0, (short)0, acc2, false, false);
        acc2 = __builtin_amdgcn_wmma_f32_16x16x4_f32(false, A1, false, S1, (short)0, acc2, false, false);
        acc2 = __builtin_amdgcn_wmma_f32_16x16x4_f32(false, A2, false, S2, (short)0, acc2, false, false);
        float t1 = tn1[ncol], t2 = tn2[ncol];
        for (int r = 0; r < 8; r++) {
            float d1 = pnr[r] + t1 - 2.0f * acc1[r];
            if (d1 < mv1[r]) { mv1[r] = d1; mi1[r] = ncol; }
            float d2 = pnr[r] + t2 - 2.0f * acc2[r];
            if (d2 < mv2[r]) { mv2[r] = d2; mi2[r] = ncol; }
        }
    }
    // reduce (min, argmin) over the 16 lanes of each half (C-layout N dimension);
    // xor masks < 16 stay within each half. Ties -> lowest index (matches jnp.argmin).
    int* rind = (int*)(wb + PB_RIND);
    for (int r = 0; r < 8; r++) {
        float v1 = mv1[r], v2 = mv2[r];
        int i1 = mi1[r], i2 = mi2[r];
        for (int m = 1; m < 16; m <<= 1) {
            float ov = __shfl_xor(v1, m, 32); int oi = __shfl_xor(i1, m, 32);
            if (ov < v1 || (ov == v1 && oi < i1)) { v1 = ov; i1 = oi; }
            float ow = __shfl_xor(v2, m, 32); int oj = __shfl_xor(i2, m, 32);
            if (ow < v2 || (ow == v2 && oj < i2)) { v2 = ow; i2 = oj; }
        }
        if (row == 0) rind[seed0 + r + 8 * half] = (v1 < v2) ? i1 : i2;
    }
}

// ---------------- K7: final scalar gathers + outputs ----------------
__global__ void k7_out(const float* labels, const float* offs, const float* dpred,
                       float* ws, float* out) {
    int b = blockIdx.y;
    int n = blockIdx.x * blockDim.x + threadIdx.x;
    if (n >= NSEED) return;
    float* wb = ws + OF_BATCH + (size_t)b * PB_SIZE;
    int ri = ((const int*)(wb + PB_RIND))[n];
    int v = ri / NA;
    int aa = ((const int*)(wb + PB_AIND))[ri];
    int g = ((const int*)(wb + PB_NN))[n];
    int vv = ((const int*)(wb + PB_VIND))[v];
    float dpv = dpred[b * NSEED + n];
    int di = 0; float bd = 3.4e38f;
    for (int d = 0; d < ND; d++) {
        float dd = fabsf(dpv - (0.01f + 0.01f * d));
        if (dd < bd) { bd = dd; di = d; }
    }
    size_t lidx = ((((size_t)(b * NGP + g)) * NV + vv) * NA + aa) * ND + di;
    float gs = labels[lidx];
    float gw = offs[lidx];
    float gd = 0.01f + 0.01f * di;
    const float* T = wb + PB_TR + (size_t)ri * 9;
    const float* pt = wb + PB_GPT + g * 3;
    bool mask = (gs > 0.0f) && (gw <= 0.1f);
    float sco = mask ? (1.1f - gs) : 0.0f;
    const float wbins[4] = {0.02f, 0.04f, 0.06f, 0.08f};
    const float sbins[9] = {0.1f, 0.2f, 0.3f, 0.4f, 0.5f, 0.6f, 0.7f, 0.8f, 0.9f};
    int wid = 0; for (int i = 0; i < 4; i++) wid += (wbins[i] < gw) ? 1 : 0;
    int sid = 0; for (int i = 0; i < 9; i++) sid += (sbins[i] < sco) ? 1 : 0;
    const int BN = NBATCH * NSEED;
    int bn = b * NSEED + n;
    float* pts_o = out;
    float* rot_o = out + (size_t)BN * 3;
    float* sc_o  = out + (size_t)BN * 9;
    float* wd_o  = out + (size_t)BN * 10;
    float* dp_o  = out + (size_t)BN * 11;
    int* wid_o   = (int*)(out + (size_t)BN * 12);
    int* sid_o   = (int*)(out + (size_t)BN * 13);
    pts_o[bn * 3 + 0] = pt[0]; pts_o[bn * 3 + 1] = pt[1]; pts_o[bn * 3 + 2] = pt[2];
    for (int i = 0; i < 6; i++) rot_o[bn * 6 + i] = T[i];  // rows 0,1 of tR
    sc_o[bn] = sco; wd_o[bn] = gw; dp_o[bn] = gd;
    wid_o[bn] = wid; sid_o[bn] = sid;
}

extern "C" void kernel_launch(void* const* d_in, const int* in_sizes, int n_in,
                              void* d_out, int out_size, void* d_ws, size_t ws_size,
                              hipStream_t stream) {
    (void)in_sizes; (void)n_in; (void)out_size; (void)ws_size;
    const float* pc     = (const float*)d_in[0];
    const float* pose   = (const float*)d_in[1];
    const float* gp     = (const float*)d_in[2];
    const float* labels = (const float*)d_in[3];
    const float* offs   = (const float*)d_in[4];
    const float* r6     = (const float*)d_in[5];
    const float* dpred  = (const float*)d_in[6];
    float* out = (float*)d_out;
    float* ws  = (float*)d_ws;

    k0_const<<<(NM + 255) / 256, 256, 0, stream>>>(ws);
    k1_viewinds<<<NBATCH, 320, 0, stream>>>(pose, ws);
    k2_align<<<dim3((NM + 255) / 256, NBATCH), 256, 0, stream>>>(pose, ws);
    k3_gpt<<<dim3(NGP / 256, NBATCH), 256, 0, stream>>>(pose, gp, ws);
    k4_nn<<<dim3(NSEED / 256, NBATCH), 256, 0, stream>>>(pc, ws);
    k5_pbase<<<dim3(NSEED / 256, NBATCH), 256, 0, stream>>>(r6, ws);
    k6_argmin<<<dim3(NSEED / 16, NBATCH), 32, 0, stream>>>(ws);
    k7_out<<<dim3(NSEED / 256, NBATCH), 256, 0, stream>>>(labels, offs, dpred, ws, out);
}